// WaveFieldRenderer_68410239090720
// MI455X (gfx1250) — compile-verified
//
#include <hip/hip_runtime.h>
#include <hip/hip_bf16.h>
#include <math.h>

// ---------------- problem constants (from reference) ----------------
constexpr int   kH = 800, kW = 800;
constexpr float kFX = 800.f, kFY = 800.f, kCX = 400.f, kCY = 400.f;
constexpr float kNEAR = 0.01f, kFAR = 100.f;
constexpr float kMAXR = 16.f;
constexpr int   kN = 30000;

typedef __attribute__((ext_vector_type(2))) float v2f;
typedef __attribute__((ext_vector_type(8))) float v8f;

// ---------------------------------------------------------------
// Kernel 1: per-gaussian preprocessing.
// params[g*16]: {u, v, ia, ibc, id, xlo, xhi, ylo, yhi, pad, w0..w5}
//   w[c] = opacity*color[c%3]*cos(phase) (c<3) / *sin(phase) (c>=3)
// bounds[g] = float4(xlo, xhi, ylo, yhi)  (inverted sentinel if invisible)
// ---------------------------------------------------------------
__global__ __launch_bounds__(256)
void prep_kernel(const float* __restrict__ pos, const float* __restrict__ scl,
                 const float* __restrict__ rot, const float* __restrict__ col,
                 const float* __restrict__ opa, const float* __restrict__ pha,
                 const float* __restrict__ vm,
                 float* __restrict__ params, float4* __restrict__ bounds)
{
    int g = blockIdx.x * blockDim.x + threadIdx.x;
    if (g >= kN) return;

    float V[12];
#pragma unroll
    for (int i = 0; i < 12; ++i) V[i] = vm[i];   // top 3 rows of view matrix

    float p0 = pos[3*g], p1 = pos[3*g+1], p2 = pos[3*g+2];
    float x = V[0]*p0 + V[1]*p1 + V[2]*p2  + V[3];
    float y = V[4]*p0 + V[5]*p1 + V[6]*p2  + V[7];
    float z = V[8]*p0 + V[9]*p1 + V[10]*p2 + V[11];
    float depth = -z;

    // quaternion -> rotation
    float qw = rot[4*g], qx = rot[4*g+1], qy = rot[4*g+2], qz = rot[4*g+3];
    float qn = fmaxf(sqrtf(qw*qw + qx*qx + qy*qy + qz*qz), 1e-12f);
    float qi = 1.f / qn; qw *= qi; qx *= qi; qy *= qi; qz *= qi;
    float R[9];
    R[0] = 1.f - 2.f*(qy*qy + qz*qz); R[1] = 2.f*(qx*qy - qw*qz); R[2] = 2.f*(qx*qz + qw*qy);
    R[3] = 2.f*(qx*qy + qw*qz); R[4] = 1.f - 2.f*(qx*qx + qz*qz); R[5] = 2.f*(qy*qz - qw*qx);
    R[6] = 2.f*(qx*qz - qw*qy); R[7] = 2.f*(qy*qz + qw*qx); R[8] = 1.f - 2.f*(qx*qx + qy*qy);

    // R_cam = V3x3 @ R, then scale columns
    float M[9];
#pragma unroll
    for (int i = 0; i < 3; ++i)
#pragma unroll
        for (int j = 0; j < 3; ++j)
            M[i*3+j] = V[i*4+0]*R[0*3+j] + V[i*4+1]*R[1*3+j] + V[i*4+2]*R[2*3+j];
    float s0 = scl[3*g], s1 = scl[3*g+1], s2 = scl[3*g+2];
#pragma unroll
    for (int i = 0; i < 3; ++i) { M[i*3+0] *= s0; M[i*3+1] *= s1; M[i*3+2] *= s2; }

    // cov3d = M M^T (symmetric)
    float c00 = M[0]*M[0] + M[1]*M[1] + M[2]*M[2];
    float c01 = M[0]*M[3] + M[1]*M[4] + M[2]*M[5];
    float c02 = M[0]*M[6] + M[1]*M[7] + M[2]*M[8];
    float c11 = M[3]*M[3] + M[4]*M[4] + M[5]*M[5];
    float c12 = M[3]*M[6] + M[4]*M[7] + M[5]*M[8];
    float c22 = M[6]*M[6] + M[7]*M[7] + M[8]*M[8];

    float zcl = fmaxf(fabsf(z), 0.01f);
    float zs  = z + 1e-8f;
    float sgn = (zs > 0.f) ? 1.f : ((zs < 0.f) ? -1.f : 0.f);
    float z_safe = zcl * sgn;
    float z2 = z_safe * z_safe;

    float j00 = kFX / (-z_safe);
    float j02 = kFX * x / z2;
    float j11 = kFY / z_safe;
    float j12 = kFY * y / z2;

    // cov2d = J C J^T
    float v00 = j00*c00 + j02*c02;
    float v01 = j00*c01 + j02*c12;
    float v02 = j00*c02 + j02*c22;
    float v10 = j11*c01 + j12*c02;
    float v11 = j11*c11 + j12*c12;
    float v12 = j11*c12 + j12*c22;
    float a  = v00*j00 + v02*j02;
    float b  = v01*j11 + v02*j12;
    float cc = v10*j00 + v12*j02;
    float d  = v11*j11 + v12*j12;

    float u  = kFX * x / (-z_safe) + kCX;
    float vv = kFY * (-y) / (-z_safe) + kCY;

    float trace = a + d;
    float det  = fmaxf(a*d - b*cc, 1e-6f);
    float disc = fmaxf(trace*trace - 4.f*det, 0.f);
    float meig = 0.5f * (trace + sqrtf(disc));
    float rad  = fminf(3.f * sqrtf(fmaxf(meig, 1e-6f)), kMAXR);

    bool vis = (depth > kNEAR) && (depth < kFAR) &&
               (u + rad > 0.f) && (u - rad < (float)kW) &&
               (vv + rad > 0.f) && (vv - rad < (float)kH);

    float ra = a + 1e-4f, rd = d + 1e-4f;
    float rdet = ra*rd - b*cc;
    float ia  = rd / rdet;
    float ibc = (-b - cc) / rdet;
    float idd = ra / rdet;

    float xlo = floorf(u - rad),  xhi = floorf(u + rad);
    float ylo = floorf(vv - rad), yhi = floorf(vv + rad);
    if (!vis) { xlo = 1e9f; xhi = -1e9f; ylo = 1e9f; yhi = -1e9f; }

    float ph = pha[g], op = opa[g];
    float cp = cosf(ph), sp = sinf(ph);
    float c0 = col[3*g], c1 = col[3*g+1], c2 = col[3*g+2];

    float* P = params + (size_t)g * 16;
    P[0] = u;   P[1] = vv;  P[2] = ia;  P[3] = ibc;
    P[4] = idd; P[5] = xlo; P[6] = xhi; P[7] = ylo;
    P[8] = yhi; P[9] = 0.f;
    P[10] = op*c0*cp; P[11] = op*c1*cp; P[12] = op*c2*cp;
    P[13] = op*c0*sp; P[14] = op*c1*sp; P[15] = op*c2*sp;
    bounds[g] = make_float4(xlo, xhi, ylo, yhi);
}

__device__ __forceinline__ float gauss_amp(float pxf, float pyf,
                                           float u, float v, float ia, float ibc, float idd,
                                           float xlo, float xhi, float ylo, float yhi,
                                           bool valid)
{
    float dx = pxf - u, dy = pyf - v;
    float mh = ia*dx*dx + ibc*dx*dy + idd*dy*dy;
    float e  = __expf(-0.5f * mh);
    bool in  = valid && (pxf >= xlo) && (pxf <= xhi) && (pyf >= ylo) && (pyf <= yhi);
    return in ? e : 0.f;
}

// ---------------------------------------------------------------
// Kernel 2: one workgroup per 16x16 pixel tile. 8 waves (wave32), each
// wave owns two pixel rows, accumulating a 16(pixel)x16(channel) f32 D
// matrix per row via V_WMMA_F32_16X16X4_F32 with K = 4 gaussians/step.
// Gaussians are culled per tile with a deterministic ballot-compaction
// and staged into LDS. No global atomics anywhere.
// ---------------------------------------------------------------
__global__ __launch_bounds__(256)
void render_kernel(const float* __restrict__ params,
                   const float4* __restrict__ bounds,
                   float* __restrict__ out)
{
    __shared__ __align__(16) float sP[256 * 16];   // staged survivor params
    __shared__ int sIdx[256];                      // survivor gaussian indices
    __shared__ int sWave[9];                       // per-wave scan + total

    const int t    = threadIdx.x;
    const int lane = t & 31;
    const int wave = t >> 5;
    const int tx   = blockIdx.x, ty = blockIdx.y;

    const float tX0 = (float)(tx * 16),      tY0 = (float)(ty * 16);
    const float tX1 = (float)(tx * 16 + 15), tY1 = (float)(ty * 16 + 15);

    const int   m    = lane & 15;        // pixel-x within tile == A-matrix M row
    const int   kh   = lane >> 4;        // which K pair this lane supplies
    const int   ch   = m;                // channel column for B / D
    const float pxf  = tX0 + (float)m;
    const float py0f = tY0 + (float)(2 * wave);
    const float py1f = py0f + 1.f;

    v8f d0 = {}; v8f d1 = {};            // register accumulators (pixels x channels)

    const int nChunks = (kN + 255) / 256;
    for (int c = 0; c < nChunks; ++c) {
        // ---- tile-level cull over this chunk of 256 gaussians ----
        int g = c * 256 + t;
        bool flag = false;
        if (g < kN) {
            float4 bb = bounds[g];   // xlo,xhi,ylo,yhi
            flag = (bb.x <= tX1) && (bb.y >= tX0) && (bb.z <= tY1) && (bb.w >= tY0);
        }
        unsigned bm  = (unsigned)__ballot(flag);
        int      pre = __popc(bm & ((1u << lane) - 1u));
        if (lane == 0) sWave[wave] = __popc(bm);
        __syncthreads();
        if (t == 0) {                      // tiny 8-element exclusive scan
            int s = 0;
#pragma unroll
            for (int i = 0; i < 8; ++i) { int n = sWave[i]; sWave[i] = s; s += n; }
            sWave[8] = s;
        }
        __syncthreads();
        int cnt = sWave[8];
        if (flag) sIdx[sWave[wave] + pre] = g;   // deterministic order
        __syncthreads();

        if (cnt > 0) {
            // ---- stage survivor params into LDS (4 x float4 each) ----
            for (int j = t; j < cnt * 4; j += 256) {
                int gi = sIdx[j >> 2];
                ((float4*)sP)[j] = ((const float4*)(params + (size_t)gi * 16))[j & 3];
            }
            __syncthreads();

            // ---- WMMA accumulation: K = 4 gaussians per step ----
            int ng = (cnt + 3) >> 2;
            for (int q = 0; q < ng; ++q) {
                int ga = q * 4 + kh * 2;
                int gb = ga + 1;
                bool va = ga < cnt, vb = gb < cnt;
                const float* Pa = sP + (va ? ga : 0) * 16;
                const float* Pb = sP + (vb ? gb : 0) * 16;

                float4 a0 = *(const float4*)(Pa + 0);   // u,v,ia,ibc
                float4 a4 = *(const float4*)(Pa + 4);   // id,xlo,xhi,ylo
                float  ay = Pa[8];
                float4 b0 = *(const float4*)(Pb + 0);
                float4 b4 = *(const float4*)(Pb + 4);
                float  by = Pb[8];

                v2f A0, A1, B;
                A0[0] = gauss_amp(pxf, py0f, a0.x, a0.y, a0.z, a0.w, a4.x, a4.y, a4.z, a4.w, ay, va);
                A0[1] = gauss_amp(pxf, py0f, b0.x, b0.y, b0.z, b0.w, b4.x, b4.y, b4.z, b4.w, by, vb);
                A1[0] = gauss_amp(pxf, py1f, a0.x, a0.y, a0.z, a0.w, a4.x, a4.y, a4.z, a4.w, ay, va);
                A1[1] = gauss_amp(pxf, py1f, b0.x, b0.y, b0.z, b0.w, b4.x, b4.y, b4.z, b4.w, by, vb);

                int wc = (ch < 6) ? ch : 5;             // clamp LDS index, mask below
                float wa = Pa[10 + wc], wb = Pb[10 + wc];
                B[0] = (ch < 6 && va) ? wa : 0.f;
                B[1] = (ch < 6 && vb) ? wb : 0.f;

                // D(16px x 16ch) += A(16px x 4g) * B(4g x 16ch)
                d0 = __builtin_amdgcn_wmma_f32_16x16x4_f32(false, A0, false, B,
                                                           (short)0, d0, false, false);
                d1 = __builtin_amdgcn_wmma_f32_16x16x4_f32(false, A1, false, B,
                                                           (short)0, d1, false, false);
            }
        }
        __syncthreads();
    }

    // ---- |field|^2 and store: pair channel c (real) with c+3 (imag) ----
    const int half    = lane >> 4;
    const int srcLane = (lane & 16) + ch + 3;   // same half, imag channel
    const int py0     = ty * 16 + 2 * wave;
#pragma unroll
    for (int j = 0; j < 8; ++j) {
        float wi0 = __shfl(d0[j], srcLane, 32);   // executed by all lanes
        float wi1 = __shfl(d1[j], srcLane, 32);
        if (ch < 3) {
            int px = tx * 16 + (half ? j + 8 : j);
            float wr0 = d0[j], wr1 = d1[j];
            out[(size_t)ch * kW * kH + (size_t)py0 * kW + px]       = wr0*wr0 + wi0*wi0;
            out[(size_t)ch * kW * kH + (size_t)(py0 + 1) * kW + px] = wr1*wr1 + wi1*wi1;
        }
    }
}

extern "C" void kernel_launch(void* const* d_in, const int* in_sizes, int n_in,
                              void* d_out, int out_size, void* d_ws, size_t ws_size,
                              hipStream_t stream)
{
    const float* pos = (const float*)d_in[0];
    const float* scl = (const float*)d_in[1];
    const float* rot = (const float*)d_in[2];
    const float* col = (const float*)d_in[3];
    const float* opa = (const float*)d_in[4];
    const float* pha = (const float*)d_in[5];
    const float* vm  = (const float*)d_in[6];
    float* out = (float*)d_out;

    float*  params = (float*)d_ws;                       // kN*16 floats (1.92 MB)
    float4* bnds   = (float4*)(params + (size_t)kN*16);  // kN float4   (0.48 MB)

    prep_kernel<<<(kN + 255) / 256, 256, 0, stream>>>(pos, scl, rot, col, opa, pha, vm,
                                                      params, bnds);
    dim3 grid(kW / 16, kH / 16);
    render_kernel<<<grid, 256, 0, stream>>>(params, bnds, out);
}